// CombinedLoss_45037027066266
// MI455X (gfx1250) — compile-verified
//
#include <hip/hip_runtime.h>
#include <math.h>

#define EPSILON 0.1f
#define MARGIN  0.3f

typedef __attribute__((ext_vector_type(2)))  float  v2f;
typedef __attribute__((ext_vector_type(4)))  float  v4f;
typedef __attribute__((ext_vector_type(8)))  float  v8f;

// ---------------------------------------------------------------- utilities
__device__ __forceinline__ void online_lse(float v, float& m, float& s) {
    // streaming logsumexp: running max m, running sum s of exp(x-m)
    if (v > m) { s = s * __expf(m - v) + 1.0f; m = v; }
    else       { s += __expf(v - m); }
}

__global__ void init_accum_kernel(float* acc) { acc[0] = 0.0f; acc[1] = 0.0f; }

// ---------------------------------------------------------- CE label smooth
// one block (256 thr = 8 waves) per row; single pass over C classes.
// Non-temporal b128 loads: the 410MB cls_score is touched once — keep it
// from evicting the feature/dist working set (12MB) out of the 192MB L2.
// loss_row = lse - (1-eps)*x[label] - (eps/C)*sum(x)
__global__ __launch_bounds__(256)
void ce_ls_kernel(const float* __restrict__ x, const int* __restrict__ labels,
                  float* __restrict__ accum, int N, int C) {
    const int row = blockIdx.x;
    const int tid = threadIdx.x;
    const float* xr = x + (size_t)row * C;

    float m = -INFINITY, s = 0.0f, sumx = 0.0f;
    const int C4 = C >> 2;
    const v4f* xr4 = reinterpret_cast<const v4f*>(xr);
    for (int c = tid; c < C4; c += 256) {
        v4f v = __builtin_nontemporal_load(&xr4[c]);
        sumx += v.x + v.y + v.z + v.w;
        online_lse(v.x, m, s); online_lse(v.y, m, s);
        online_lse(v.z, m, s); online_lse(v.w, m, s);
    }
    for (int c = (C4 << 2) + tid; c < C; c += 256) {   // tail (C%4 == 0 here)
        float v = xr[c]; sumx += v; online_lse(v, m, s);
    }

    __shared__ float sm[256], ss[256], sx[256];
    sm[tid] = m; ss[tid] = s; sx[tid] = sumx;
    __syncthreads();
    for (int off = 128; off > 0; off >>= 1) {
        if (tid < off) {
            float m1 = sm[tid], s1 = ss[tid];
            float m2 = sm[tid + off], s2 = ss[tid + off];
            float mm = fmaxf(m1, m2);
            ss[tid] = s1 * __expf(m1 - mm) + s2 * __expf(m2 - mm);
            sm[tid] = mm;
            sx[tid] += sx[tid + off];
        }
        __syncthreads();
    }
    if (tid == 0) {
        float lse = sm[0] + __logf(ss[0]);
        float xl  = xr[labels[row]];
        float loss_row = lse - (1.0f - EPSILON) * xl - (EPSILON / (float)C) * sx[0];
        atomicAdd(&accum[0], loss_row / (float)N);
    }
}

// ------------------------------------------------------------ squared norms
__global__ __launch_bounds__(256)
void sqnorm_kernel(const float* __restrict__ X, float* __restrict__ sq, int D) {
    const int row = blockIdx.x;
    const int tid = threadIdx.x;
    const float* xr = X + (size_t)row * D;
    float acc = 0.0f;
    for (int c = tid; c < D; c += 256) { float v = xr[c]; acc += v * v; }
    __shared__ float sh[256];
    sh[tid] = acc;
    __syncthreads();
    for (int off = 128; off > 0; off >>= 1) {
        if (tid < off) sh[tid] += sh[tid + off];
        __syncthreads();
    }
    if (tid == 0) sq[row] = sh[0];
}

// -------------------------------------------------- gram + distance (WMMA)
// One wave32 per 32x32 output tile of G = X X^T, register-blocked 2x2 over
// 16x16 WMMA fragments: 4 global_load_b64 feed 4 v_wmma_f32_16x16x4_f32 per
// k-step (8 FLOP/byte from cache, 2x the naive 16x16 tiling).
// A layout (32-bit 16x4): lane l<16 -> row l, K = 2*hi+{0,1} per float2;
// B (4x16) per-lane data is the identical pattern with the column base
// (B = X^T).  Epilogue fuses dist = sqrt(clip(sq_i + sq_j - 2g, 1e-12)).
__global__ __launch_bounds__(32)
void gram_dist_kernel(const float* __restrict__ X, const float* __restrict__ sq,
                      float* __restrict__ dist, int N, int D) {
    const int lane = threadIdx.x & 31;
    const int mrow = lane & 15;
    const int hi   = lane >> 4;
    const int rowA = blockIdx.x * 32;
    const int rowB = blockIdx.y * 32;

    const float* pa0 = X + (size_t)(rowA      + mrow) * D + 2 * hi;
    const float* pa1 = X + (size_t)(rowA + 16 + mrow) * D + 2 * hi;
    const float* pb0 = X + (size_t)(rowB      + mrow) * D + 2 * hi;
    const float* pb1 = X + (size_t)(rowB + 16 + mrow) * D + 2 * hi;

    v8f c00 = {}, c01 = {}, c10 = {}, c11 = {};
    #pragma unroll 2
    for (int k0 = 0; k0 < D; k0 += 4) {
        v2f a0 = *reinterpret_cast<const v2f*>(pa0 + k0);
        v2f a1 = *reinterpret_cast<const v2f*>(pa1 + k0);
        v2f b0 = *reinterpret_cast<const v2f*>(pb0 + k0);
        v2f b1 = *reinterpret_cast<const v2f*>(pb1 + k0);
        c00 = __builtin_amdgcn_wmma_f32_16x16x4_f32(false, a0, false, b0, (short)0, c00, false, false);
        c01 = __builtin_amdgcn_wmma_f32_16x16x4_f32(false, a0, false, b1, (short)0, c01, false, false);
        c10 = __builtin_amdgcn_wmma_f32_16x16x4_f32(false, a1, false, b0, (short)0, c10, false, false);
        c11 = __builtin_amdgcn_wmma_f32_16x16x4_f32(false, a1, false, b1, (short)0, c11, false, false);
    }

    // epilogue: C/D layout VGPR r -> M = r (+8 for high half-lanes), N = lane&15
    v8f acc[2][2] = {{c00, c01}, {c10, c11}};
    #pragma unroll
    for (int ti = 0; ti < 2; ++ti) {
        #pragma unroll
        for (int tj = 0; tj < 2; ++tj) {
            const int col = rowB + 16 * tj + mrow;
            const float sqc = sq[col];
            #pragma unroll
            for (int r = 0; r < 8; ++r) {
                const int row = rowA + 16 * ti + r + 8 * hi;
                float d = sq[row] + sqc - 2.0f * acc[ti][tj][r];
                d = sqrtf(fmaxf(d, 1e-12f));
                dist[(size_t)row * N + col] = d;
            }
        }
    }
}

// ------------------------------------------- batch-hard mining per row
__global__ __launch_bounds__(256)
void rowred_kernel(const float* __restrict__ dist, const int* __restrict__ labels,
                   float* __restrict__ accum, int N) {
    const int row = blockIdx.x;
    const int tid = threadIdx.x;
    const int li  = labels[row];
    const float* dr = dist + (size_t)row * N;
    float mp = -INFINITY, mn = INFINITY;
    for (int j = tid; j < N; j += 256) {
        float d = dr[j];
        if (labels[j] == li) mp = fmaxf(mp, d);
        else                 mn = fminf(mn, d);
    }
    __shared__ float sp[256], sn[256];
    sp[tid] = mp; sn[tid] = mn;
    __syncthreads();
    for (int off = 128; off > 0; off >>= 1) {
        if (tid < off) {
            sp[tid] = fmaxf(sp[tid], sp[tid + off]);
            sn[tid] = fminf(sn[tid], sn[tid + off]);
        }
        __syncthreads();
    }
    if (tid == 0) {
        float t = fmaxf(sp[0] - sn[0] + MARGIN, 0.0f);
        atomicAdd(&accum[1], t / (float)N);
    }
}

// ------------------------------------------------------------------ output
__global__ void finalize_kernel(const float* __restrict__ accum, float* __restrict__ out) {
    out[0] = accum[0] + accum[1];   // total loss
    out[1] = accum[0];              // id loss
    out[2] = accum[1];              // triplet loss
}

// ------------------------------------------------------------------ launch
extern "C" void kernel_launch(void* const* d_in, const int* in_sizes, int n_in,
                              void* d_out, int out_size, void* d_ws, size_t ws_size,
                              hipStream_t stream) {
    const float* cls_score   = (const float*)d_in[0];  // [N, C]
    const float* global_feat = (const float*)d_in[1];  // [N, D]
    // d_in[2] = feat : unused by the reference loss
    const int*   labels      = (const int*)d_in[3];    // [N]

    const int N = in_sizes[3];
    const int C = in_sizes[0] / N;
    const int D = in_sizes[1] / N;

    // workspace layout
    float* sq    = (float*)d_ws;                        // N floats (@0)
    float* accum = (float*)((char*)d_ws + 4096);        // [id, tri]
    float* dist  = (float*)((char*)d_ws + 8192);        // N*N floats
    float* out   = (float*)d_out;

    init_accum_kernel<<<1, 1, 0, stream>>>(accum);
    ce_ls_kernel<<<N, 256, 0, stream>>>(cls_score, labels, accum, N, C);
    sqnorm_kernel<<<N, 256, 0, stream>>>(global_feat, sq, D);
    gram_dist_kernel<<<dim3(N / 32, N / 32), 32, 0, stream>>>(global_feat, sq, dist, N, D);
    rowred_kernel<<<N, 256, 0, stream>>>(dist, labels, accum, N);
    finalize_kernel<<<1, 1, 0, stream>>>(accum, out);
}